// LovaszSoftmaxLoss_4020089389750
// MI455X (gfx1250) — compile-verified
//
#include <hip/hip_runtime.h>
#include <stdint.h>

#define NCLS    19
#define HW_     (512*1024)
#define NPIX    (4*HW_)             // 2,097,152 pixels
#define TILE    4096
#define NTILES  (NPIX/TILE)         // 512
#define THREADS 256
#define ITEMS   16
#define NHIST   (16*NTILES)         // 8192 per class

typedef unsigned int tdm_u4 __attribute__((ext_vector_type(4)));
typedef int          tdm_i8 __attribute__((ext_vector_type(8)));
typedef int          tdm_i4 __attribute__((ext_vector_type(4)));
typedef float        v2f    __attribute__((ext_vector_type(2)));
typedef float        v8f    __attribute__((ext_vector_type(8)));

// padded LDS layout produced by TDM pad feature: +1 dword every 16 dwords
#define LDSPAD(j) ((j) + ((j) >> 4))
#define TILE_PADDED (TILE + TILE/16)

// ---------------------------------------------------------------------------
// TDM: DMA one 16KB tile of keys into LDS with padding (1 dword per 16).
// Issued by wave 0 only; caller must __syncthreads() after.
// ---------------------------------------------------------------------------
__device__ __forceinline__ void tile_to_lds(const uint32_t* __restrict__ gsrc,
                                            uint32_t* lds)
{
#if __has_builtin(__builtin_amdgcn_tensor_load_to_lds) && __has_builtin(__builtin_amdgcn_s_wait_tensorcnt)
  if (threadIdx.x < 32) {   // wave-uniform guard: only wave 0 issues the DMA
    uint64_t ga = (uint64_t)(uintptr_t)gsrc;
    uint32_t lo = (uint32_t)(uintptr_t)lds;      // LDS byte offset
    tdm_u4 g0;
    g0[0] = 1u;                                   // count=1, user descriptor
    g0[1] = lo;                                   // lds_addr
    g0[2] = (uint32_t)ga;                         // global_addr[31:0]
    g0[3] = (uint32_t)((ga >> 32) & 0x01FFFFFFu)  // global_addr[56:32]
            | (2u << 30);                         // type=2 (image)
    tdm_i8 g1;
    // data_size=4B, pad_enable, pad_interval=3 (16 dwords), pad_amount=0 (1 dword)
    g1[0] = (int)((2u << 16) | (1u << 20) | (3u << 22));
    g1[1] = (int)(((uint32_t)TILE & 0xFFFFu) << 16); // tensor_dim0 lo16
    g1[2] = (int)(1u << 16);                         // tensor_dim1 = 1
    g1[3] = (int)((uint32_t)TILE << 16);             // tile_dim0 = 4096
    g1[4] = 1;                                       // tile_dim1 = 1
    g1[5] = TILE;                                    // tensor_dim0_stride
    g1[6] = 0;
    g1[7] = 0;
    tdm_i4 z4; z4[0]=0; z4[1]=0; z4[2]=0; z4[3]=0;   // groups 2/3 unused (<=2D)
    tdm_i8 z8; z8[0]=0; z8[1]=0; z8[2]=0; z8[3]=0;
    z8[4]=0; z8[5]=0; z8[6]=0; z8[7]=0;
    __builtin_amdgcn_tensor_load_to_lds(g0, g1, z4, z4, z8, 0);
    __builtin_amdgcn_s_wait_tensorcnt(0);
  }
#else
  for (int j = (int)threadIdx.x; j < TILE; j += THREADS)
    lds[LDSPAD(j)] = gsrc[j];
#endif
}

// ---------------------------------------------------------------------------
// Block-wide exclusive scan of 256 per-thread sums via WMMA.
// View the 256 sums as 16x16 row-major M. Then
//   P = M x U          (U upper-tri ones -> within-row inclusive prefix)
//   incl = P + L x T   (L strictly-lower ones, T[r][:] = P[r][15] broadcast)
// 8 chained V_WMMA_F32_16X16X4_F32 executed by wave 0 (EXEC all ones).
// Sums <= 2^21 so f32 accumulation is exact.
// Leaves tot[] holding the INCLUSIVE scan (tot[255] = grand total).
// ---------------------------------------------------------------------------
__device__ __forceinline__ uint32_t block_excl_scan_256(uint32_t mySum,
                                                        float* tot)
{
  int tid = threadIdx.x;
  tot[tid] = (float)mySum;
  __syncthreads();
#if __has_builtin(__builtin_amdgcn_wmma_f32_16x16x4_f32)
  if (tid < 32) {                       // wave-uniform: whole wave 0, EXEC=~0
    const int half = tid >> 4;          // 0: K/M low half, 1: high half
    const int r    = tid & 15;          // A row / B,D column
    v8f acc = {};
    // ---- P = M x U ----
#pragma unroll
    for (int k = 0; k < 4; k++) {
      const int K0 = 4 * k + 2 * half;  // VGPR0 K index (A & B layouts)
      const int K1 = K0 + 1;            // VGPR1 K index
      v2f a, b;
      a[0] = tot[r * 16 + K0];          // A[r][K0]
      a[1] = tot[r * 16 + K1];          // A[r][K1]
      b[0] = (K0 <= r) ? 1.f : 0.f;     // U[K0][r]
      b[1] = (K1 <= r) ? 1.f : 0.f;     // U[K1][r]
      acc = __builtin_amdgcn_wmma_f32_16x16x4_f32(false, a, false, b,
                                                  (short)0, acc, false, false);
    }
    v8f P = acc;                        // snapshot: row totals R[m] = P[m][15]
    // R[m]: m<8 -> lane 15 vgpr m ; m>=8 -> lane 31 vgpr m-8
    // ---- incl = P + L x T ----
#pragma unroll
    for (int k = 0; k < 4; k++) {
      const int K0 = 4 * k + 2 * half;
      const int K1 = K0 + 1;
      const int rA0 = 4 * k;            // rows needed: {4k,4k+2} / {4k+1,4k+3}
      const int rA1 = 4 * k + 2;
      const int rB0 = 4 * k + 1;
      const int rB1 = 4 * k + 3;
      float q0a = __shfl(P[rA0 & 7], (rA0 < 8) ? 15 : 31, 32);
      float q0b = __shfl(P[rA1 & 7], (rA1 < 8) ? 15 : 31, 32);
      float q1a = __shfl(P[rB0 & 7], (rB0 < 8) ? 15 : 31, 32);
      float q1b = __shfl(P[rB1 & 7], (rB1 < 8) ? 15 : 31, 32);
      v2f a, b;
      a[0] = (K0 < r) ? 1.f : 0.f;      // Lstrict[r][K0]
      a[1] = (K1 < r) ? 1.f : 0.f;      // Lstrict[r][K1]
      b[0] = half ? q0b : q0a;          // T[K0][r] = R[K0]
      b[1] = half ? q1b : q1a;          // T[K1][r] = R[K1]
      acc = __builtin_amdgcn_wmma_f32_16x16x4_f32(false, a, false, b,
                                                  (short)0, acc, false, false);
    }
    // write inclusive scan back: vgpr j -> element (M = j + 8*half, N = r)
#pragma unroll
    for (int j = 0; j < 8; j++)
      tot[(j + 8 * half) * 16 + r] = acc[j];
  }
  __syncthreads();
#else
  {
    for (int off = 1; off < 256; off <<= 1) {
      float v = 0.f;
      if (tid >= off) v = tot[tid - off];
      __syncthreads();
      if (tid >= off) tot[tid] += v;
      __syncthreads();
    }
  }
#endif
  uint32_t incl = (uint32_t)tot[tid];
  return incl - mySum;
}

// ---------------------------------------------------------------------------
// Pass A: per-pixel softmax -> per-class packed sort keys.
// key = ~((bits(err)<<1) | fg)  so ascending radix == descending err.
// ---------------------------------------------------------------------------
__global__ void pass_a(const float* __restrict__ in,
                       const long long* __restrict__ tgt,
                       uint32_t* __restrict__ keys)
{
  int n = blockIdx.x * THREADS + threadIdx.x;
  long long t = tgt[n];
  bool valid = (t != 255);
  int b  = n >> 19;          // / HW_
  int hw = n & (HW_ - 1);
  const float* base = in + (size_t)b * NCLS * HW_ + hw;

  float v[NCLS];
  float m = -3.4e38f;
#pragma unroll
  for (int c = 0; c < NCLS; c++) { v[c] = base[(size_t)c * HW_]; m = fmaxf(m, v[c]); }
  float s = 0.f;
#pragma unroll
  for (int c = 0; c < NCLS; c++) { v[c] = __expf(v[c] - m); s += v[c]; }
  float inv = 1.0f / s;
  int ti = (int)t;
#pragma unroll
  for (int c = 0; c < NCLS; c++) {
    float p   = v[c] * inv;
    bool  ish = valid && (c == ti);
    float err = valid ? fabsf((ish ? 1.0f : 0.0f) - p) : 0.0f;
    uint32_t key = ~((__float_as_uint(err) << 1) | (ish ? 1u : 0u));
    keys[(size_t)c * NPIX + n] = key;
  }
}

// ---------------------------------------------------------------------------
// Radix pass 1/3: per-tile 16-bin histogram (TDM-staged tile in LDS).
// ---------------------------------------------------------------------------
__global__ void hist_kernel(const uint32_t* __restrict__ keys,
                            uint32_t* __restrict__ hist, int shift)
{
  __shared__ uint32_t smem[TILE_PADDED];
  __shared__ uint32_t h2[16 * 256];
  int cls = blockIdx.y, blk = blockIdx.x, tid = threadIdx.x;
  const uint32_t* src = keys + (size_t)cls * NPIX + (size_t)blk * TILE;

  tile_to_lds(src, smem);
  __syncthreads();

  uint32_t k[ITEMS];
#pragma unroll
  for (int i = 0; i < ITEMS; i++) {            // strided: order irrelevant for counts
    int j = i * 256 + tid;
    k[i] = smem[LDSPAD(j)];
  }
#pragma unroll
  for (int d = 0; d < 16; d++) {
    uint32_t c = 0;
#pragma unroll
    for (int i = 0; i < ITEMS; i++) c += (((k[i] >> shift) & 15u) == (uint32_t)d);
    h2[d * 256 + tid] = c;
  }
  __syncthreads();
  for (int off = 128; off > 0; off >>= 1) {
    if (tid < off) {
#pragma unroll
      for (int d = 0; d < 16; d++) h2[d * 256 + tid] += h2[d * 256 + tid + off];
    }
    __syncthreads();
  }
  if (tid < 16)
    hist[(size_t)cls * NHIST + tid * NTILES + blk] = h2[tid * 256];
}

// ---------------------------------------------------------------------------
// Radix pass 2/3: per-class exclusive scan of (digit-major, block-minor) hist.
// ---------------------------------------------------------------------------
__global__ void scan_hist(uint32_t* __restrict__ hist)
{
  __shared__ float tot[256];
  int tid = threadIdx.x;
  uint32_t* h = hist + (size_t)blockIdx.x * NHIST;
  const int CH = NHIST / 256;                  // 32
  uint32_t vals[CH];
  uint32_t s = 0;
#pragma unroll
  for (int i = 0; i < CH; i++) { vals[i] = h[tid * CH + i]; s += vals[i]; }
  uint32_t base = block_excl_scan_256(s, tot);
  uint32_t run = base;
#pragma unroll
  for (int i = 0; i < CH; i++) { uint32_t t = vals[i]; h[tid * CH + i] = run; run += t; }
}

// ---------------------------------------------------------------------------
// Radix pass 3/3: stable scatter (memory-order ranks via flat bin-major scan).
// ---------------------------------------------------------------------------
__global__ void scatter_kernel(const uint32_t* __restrict__ keysIn,
                               const uint32_t* __restrict__ hist,
                               uint32_t* __restrict__ keysOut, int shift)
{
  __shared__ uint32_t smem[TILE_PADDED];
  __shared__ uint32_t scanA[16 * 256];
  __shared__ float tot[256];
  __shared__ uint32_t gbase[16], binb[16];
  int cls = blockIdx.y, blk = blockIdx.x, tid = threadIdx.x;
  size_t clsOff = (size_t)cls * NPIX;
  const uint32_t* src = keysIn + clsOff + (size_t)blk * TILE;

  tile_to_lds(src, smem);
  __syncthreads();

  // thread-sequential items = memory order (stability); stride-17 dwords: no bank conflicts
  uint32_t k[ITEMS]; unsigned dig[ITEMS];
#pragma unroll
  for (int i = 0; i < ITEMS; i++) {
    k[i]   = smem[tid * 17 + i];               // LDSPAD(tid*16+i) = tid*17+i
    dig[i] = (k[i] >> shift) & 15u;
  }
#pragma unroll
  for (int d = 0; d < 16; d++) {
    uint32_t c = 0;
#pragma unroll
    for (int i = 0; i < ITEMS; i++) c += (dig[i] == (unsigned)d);
    scanA[d * 256 + tid] = c;
  }
  __syncthreads();

  // flat exclusive scan of 4096 counts (bin-major)
  uint32_t vals[ITEMS]; uint32_t s = 0; int base = tid * ITEMS;
#pragma unroll
  for (int i = 0; i < ITEMS; i++) { vals[i] = scanA[base + i]; s += vals[i]; }
  uint32_t ebase = block_excl_scan_256(s, tot);
  uint32_t run = ebase;
#pragma unroll
  for (int i = 0; i < ITEMS; i++) { uint32_t t = vals[i]; scanA[base + i] = run; run += t; }
  __syncthreads();

  if (tid < 16) {
    gbase[tid] = hist[(size_t)cls * NHIST + tid * NTILES + blk];
    binb[tid]  = scanA[tid * 256];
  }
  __syncthreads();

#pragma unroll
  for (int i = 0; i < ITEMS; i++) {
    unsigned d = dig[i];
    uint32_t within = 0;
#pragma unroll
    for (int i2 = 0; i2 < ITEMS; i2++)
      if (i2 < i) within += (dig[i2] == d);
    uint32_t pos = gbase[d] + (scanA[d * 256 + tid] - binb[d]) + within;
    keysOut[clsOff + pos] = k[i];
  }
}

// ---------------------------------------------------------------------------
// Gradient tail: fg block sums -> per-class scan -> partial losses -> mean.
// ---------------------------------------------------------------------------
__global__ void fg_sum(const uint32_t* __restrict__ keys, uint32_t* __restrict__ fgsum)
{
  __shared__ uint32_t red[256];
  int cls = blockIdx.y, blk = blockIdx.x, tid = threadIdx.x;
  const uint32_t* src = keys + (size_t)cls * NPIX + (size_t)blk * TILE;
  uint32_t s = 0;
  for (int j = tid; j < TILE; j += 256) s += (~src[j]) & 1u;
  red[tid] = s; __syncthreads();
  for (int off = 128; off > 0; off >>= 1) {
    if (tid < off) red[tid] += red[tid + off];
    __syncthreads();
  }
  if (tid == 0) fgsum[cls * NTILES + blk] = red[0];
}

__global__ void fg_scan(const uint32_t* __restrict__ fgsum,
                        uint32_t* __restrict__ fgbase, float* __restrict__ gts)
{
  __shared__ float tot[256];
  int tid = threadIdx.x, cls = blockIdx.x;
  const uint32_t* src = fgsum + cls * NTILES;
  uint32_t v0 = src[tid * 2], v1 = src[tid * 2 + 1];
  uint32_t ebase = block_excl_scan_256(v0 + v1, tot);
  fgbase[cls * NTILES + tid * 2]     = ebase;
  fgbase[cls * NTILES + tid * 2 + 1] = ebase + v0;
  if (tid == 0) gts[cls] = tot[255];          // tot[] holds inclusive scan; last = total
}

__global__ void loss_partial(const uint32_t* __restrict__ keys,
                             const uint32_t* __restrict__ fgbase,
                             const float* __restrict__ gts,
                             float* __restrict__ partial)
{
  __shared__ float tot[256];
  __shared__ float red[256];
  int cls = blockIdx.y, blk = blockIdx.x, tid = threadIdx.x;
  const uint4* p4 = (const uint4*)(keys + (size_t)cls * NPIX + (size_t)blk * TILE + tid * ITEMS);
  uint32_t k[ITEMS];
#pragma unroll
  for (int q = 0; q < 4; q++) {
    uint4 a = p4[q];
    k[q*4+0] = a.x; k[q*4+1] = a.y; k[q*4+2] = a.z; k[q*4+3] = a.w;
  }
  uint32_t myFg = 0;
#pragma unroll
  for (int i = 0; i < ITEMS; i++) myFg += (~k[i]) & 1u;
  uint32_t tbase = block_excl_scan_256(myFg, tot);
  uint32_t running = fgbase[cls * NTILES + blk] + tbase;   // exclusive fg count before my first item
  float g = gts[cls];
  float acc = 0.f;
  int n0 = blk * TILE + tid * ITEMS;
#pragma unroll
  for (int i = 0; i < ITEMS; i++) {
    uint32_t orig = ~k[i];
    uint32_t fg = orig & 1u;
    float err = __uint_as_float(orig >> 1);
    int n = n0 + i;
    uint32_t cumExcl = running;
    running += fg;
    uint32_t cumIncl = running;
    float uI  = g + (float)(n + 1) - (float)cumIncl;
    float jac = 1.f - (g - (float)cumIncl) / uI;
    float prevJ = 0.f;
    if (n != 0) {
      float uP = g + (float)n - (float)cumExcl;
      prevJ = 1.f - (g - (float)cumExcl) / uP;
    }
    acc += err * (jac - prevJ);
  }
  red[tid] = acc; __syncthreads();
  for (int off = 128; off > 0; off >>= 1) {
    if (tid < off) red[tid] += red[tid + off];
    __syncthreads();
  }
  if (tid == 0) partial[cls * NTILES + blk] = red[0];
}

__global__ void finalize(const float* __restrict__ partial, float* __restrict__ out)
{
  __shared__ float ls[32];
  int tid = threadIdx.x;
  float s = 0.f;
  if (tid < NCLS) {
    for (int j = 0; j < NTILES; j++) s += partial[tid * NTILES + j];   // fixed order
  }
  ls[tid] = s; __syncthreads();
  if (tid == 0) {
    float t = 0.f;
    for (int c = 0; c < NCLS; c++) t += ls[c];
    out[0] = t / (float)NCLS;
  }
}

// ---------------------------------------------------------------------------
extern "C" void kernel_launch(void* const* d_in, const int* in_sizes, int n_in,
                              void* d_out, int out_size, void* d_ws, size_t ws_size,
                              hipStream_t stream)
{
  (void)in_sizes; (void)n_in; (void)out_size; (void)ws_size;
  const float*     inp = (const float*)d_in[0];
  const long long* tgt = (const long long*)d_in[1];
  float* out = (float*)d_out;

  const size_t S_KEYS = (size_t)NCLS * NPIX * sizeof(uint32_t);   // 159.4 MB
  char* ws = (char*)d_ws;
  uint32_t* keysA  = (uint32_t*)ws;
  uint32_t* keysB  = (uint32_t*)(ws + S_KEYS);
  uint32_t* hist   = (uint32_t*)(ws + 2 * S_KEYS);                // 19*8192
  uint32_t* fgsum  = hist  + (size_t)NCLS * NHIST;
  uint32_t* fgbase = fgsum + (size_t)NCLS * NTILES;
  float*    gts    = (float*)(fgbase + (size_t)NCLS * NTILES);
  float*    part   = gts + 64;                                    // padded

  pass_a<<<NPIX / THREADS, THREADS, 0, stream>>>(inp, tgt, keysA);

  uint32_t* src = keysA; uint32_t* dst = keysB;
  for (int p = 0; p < 8; p++) {
    int shift = p * 4;
    hist_kernel   <<<dim3(NTILES, NCLS), THREADS, 0, stream>>>(src, hist, shift);
    scan_hist     <<<NCLS,               THREADS, 0, stream>>>(hist);
    scatter_kernel<<<dim3(NTILES, NCLS), THREADS, 0, stream>>>(src, hist, dst, shift);
    uint32_t* t = src; src = dst; dst = t;
  }
  // 8 passes -> sorted keys back in keysA (== src)

  fg_sum      <<<dim3(NTILES, NCLS), THREADS, 0, stream>>>(src, fgsum);
  fg_scan     <<<NCLS,               THREADS, 0, stream>>>(fgsum, fgbase, gts);
  loss_partial<<<dim3(NTILES, NCLS), THREADS, 0, stream>>>(src, fgbase, gts, part);
  finalize    <<<1, 32, 0, stream>>>(part, out);
}